// MCGRU_22127671509817
// MI455X (gfx1250) — compile-verified
//
#include <hip/hip_runtime.h>

// MC-GRU: BS=128 batch, TS=1024 steps, NF=73 independent per-feature GRUs,
// input_size=1, hidden H=8 (3H=24 gates).
//
// One wave32 per (feature, 16-row batch slab) -> 73*8 = 584 waves.
// TRANSPOSED mapping: per step compute gh^T = W_hh * h^T with
// 6x V_WMMA_F32_16X16X4_F32, A = weights (loop-invariant regs), B = data:
//   acc0 rows 0..15 = gates 0..15 (r|z): W_hh x h^T + W_ih x [x];
//        C init = b_ih+b_hh.
//   acc1 rows = n-gates DUPLICATED (row M -> gate 16+(M&7)): W_hh_n x h^T;
//        C init = b_hh_n.  Duplication makes both 16-lane halves of each
//        accumulator VGPR hold the same gh_n, so BOTH halves can compute the
//        full h update -> hreg valid in all 32 lanes.
//   acc2 = gx_n (same duplication): W_ih_n x [x]; C init = b_ih_n (separate
//        from acc1 because only gh_n is scaled by r).
//
// Layouts (ISA 7.12.2): A 16x4: lane = M(row), VGPR0 = K=0|2 (lane<16|>=16),
// VGPR1 = K=1|3.  B 4x16 (mirror): lane = N(col=batch), VGPR0 = K=0|2 row,
// VGPR1 = K=1|3.  C/D 16x16: lane = N(batch), VGPR j = row M=j (lanes<16) /
// M=j+8 (lanes>=16).
//
// Branch-free inner loop:
//   - next-step B operand: 4 v_cndmask from all-lane-valid hreg (no LDS,
//     no permutes, nothing for the compiler to sink into divergent blocks);
//   - r<->z exchange: one v_permlanex16 per row + 2 v_cndmask;
//   - output: ONE fully-active global_store_b128 per step (half0 lanes write
//     h[0:4], half1 lanes h[4:8] of the same batch row), non-temporal since
//     the 306MB output is write-once streaming.
// Gate math: sigmoid/tanh from v_exp_f32 + v_rcp_f32 (TRANS, co-execute).

#define BS 128
#define TS 1024
#define NF 73
#define H  8
#define G  24  // 3*H

typedef __attribute__((ext_vector_type(2))) float v2f;
typedef __attribute__((ext_vector_type(4))) float v4f;
typedef __attribute__((ext_vector_type(8))) float v8f;

static __device__ __forceinline__ v8f wmma_f32_k4(v2f a, v2f b, v8f c) {
  // 8 args: (neg_a, A, neg_b, B, c_mod, C, reuse_a, reuse_b)
  return __builtin_amdgcn_wmma_f32_16x16x4_f32(false, a, false, b, (short)0, c,
                                               false, false);
}

// sigmoid via v_exp_f32 + v_rcp_f32 (both TRANS, co-execute with VALU)
static __device__ __forceinline__ float fast_sigmoid(float x) {
  return __builtin_amdgcn_rcpf(1.0f + __expf(-x));
}
// tanh(x) = 1 - 2/(exp(2x)+1)
static __device__ __forceinline__ float fast_tanh(float x) {
  return fmaf(-2.0f, __builtin_amdgcn_rcpf(1.0f + __expf(2.0f * x)), 1.0f);
}

// identity swap between lanes i and i+16 (v_permlanex16_b32, sel[i] = i)
static __device__ __forceinline__ float x16(float v) {
  int i = __float_as_int(v);
  int r = __builtin_amdgcn_permlanex16(i, i, 0x76543210, (int)0xFEDCBA98,
                                       false, false);
  return __int_as_float(r);
}

__global__ __launch_bounds__(32)
void mcgru_wmma_kernel(const float* __restrict__ X,    // [BS,TS,NF]
                       const float* __restrict__ Wih,  // [NF,24,1]
                       const float* __restrict__ Whh,  // [NF,24,8]
                       const float* __restrict__ bih,  // [NF,24]
                       const float* __restrict__ bhh,  // [NF,24]
                       float* __restrict__ out) {      // [BS,TS,NF,H]
  const int lane = threadIdx.x;      // 0..31, wave32
  const int n    = lane & 15;        // batch column / weight row (mod 16)
  const int half = lane >> 4;        // K-split half of A/B operands
  const bool lo  = (half == 0);
  const int f    = blockIdx.x;       // feature
  const int b0   = blockIdx.y * 16;  // first batch row of this slab
  const int k0   = 2 * half;
  const int n7   = n & 7;            // duplicated n-gate row index

  const float* whh = Whh + f * (G * H);
  const float* wih = Wih + f * G;
  const float* bi  = bih + f * G;
  const float* bh  = bhh + f * G;

  // ---- loop-invariant A operands: weights in 16x4 A layout (lane = row) ----
  v2f at0c0, at0c1, at1c0, at1c1, awx0, awx1;
  at0c0.x = whh[n * H + k0];       at0c0.y = whh[n * H + k0 + 1];      // gates 0..15
  at0c1.x = whh[n * H + 4 + k0];   at0c1.y = whh[n * H + 4 + k0 + 1];
  // n-gate tiles: rows 0..7 AND 8..15 both carry gates 16..23 (duplicated)
  at1c0.x = whh[(16 + n7) * H + k0];     at1c0.y = whh[(16 + n7) * H + k0 + 1];
  at1c1.x = whh[(16 + n7) * H + 4 + k0]; at1c1.y = whh[(16 + n7) * H + 4 + k0 + 1];
  awx0.x = lo ? wih[n] : 0.0f;        awx0.y = 0.0f;  // A col K=0 = W_ih (r|z)
  awx1.x = lo ? wih[16 + n7] : 0.0f;  awx1.y = 0.0f;  // A col K=0 = W_ih_n

  // ---- bias accumulator seeds: C[M][*] = bias[row M], M = j + 8*half ----
  float c0s[8], c1s[8], c2s[8];
#pragma unroll
  for (int j = 0; j < 8; ++j) {
    c0s[j] = bi[j + 8 * half] + bh[j + 8 * half];  // r|z bias (rows distinct)
    c1s[j] = bh[16 + j];                           // gh_n bias (rows duplicated)
    c2s[j] = bi[16 + j];                           // gx_n bias (rows duplicated)
  }

  // ---- h state: EVERY lane n holds h[b0+n][0..7] in hreg[0..7] ----
  float hreg[8];
#pragma unroll
  for (int j = 0; j < 8; ++j) hreg[j] = 0.0f;

  const int xoff  = (b0 + n) * TS * NF + f;  // lane n: x[b0+n, t, f]
  // store base: half0 lanes write h[0:4], half1 lanes h[4:8] of batch b0+n
  const int obase = (((b0 + n) * TS) * NF + f) * H + 4 * half;

  float xcur = X[xoff];  // t = 0
  for (int t = 0; t < TS; ++t) {
    // software-pipelined x load for t+1
    const int tn = (t + 1 < TS) ? (t + 1) : (TS - 1);
    const float xnext = X[xoff + tn * NF];

    // ---- B operands: h^T in 4x16 B layout, pure cndmask from hreg.
    // chunk0: v0 = h[n][0]|h[n][2], v1 = h[n][1]|h[n][3]; chunk1: +4.
    v2f bh0, bh1, bx;
    bh0.x = lo ? hreg[0] : hreg[2];
    bh0.y = lo ? hreg[1] : hreg[3];
    bh1.x = lo ? hreg[4] : hreg[6];
    bh1.y = lo ? hreg[5] : hreg[7];
    bx.x  = lo ? xcur : 0.0f;  // B row K=0 = x[b]; K=1..3 zero
    bx.y  = 0.0f;

    v8f acc0, acc1, acc2;
#pragma unroll
    for (int j = 0; j < 8; ++j) {
      acc0[j] = c0s[j]; acc1[j] = c1s[j]; acc2[j] = c2s[j];
    }

    acc0 = wmma_f32_k4(at0c0, bh0, acc0);  // r|z: W_hh[:,0:4] x h[0:4]
    acc0 = wmma_f32_k4(at0c1, bh1, acc0);  // r|z: W_hh[:,4:8] x h[4:8]
    acc0 = wmma_f32_k4(awx0,  bx,  acc0);  // r|z: + W_ih * x
    acc1 = wmma_f32_k4(at1c0, bh0, acc1);  // gh_n: W_hh_n x h[0:4]
    acc1 = wmma_f32_k4(at1c1, bh1, acc1);  // gh_n: W_hh_n x h[4:8]
    acc2 = wmma_f32_k4(awx1,  bx,  acc2);  // gx_n = W_ih_n * x + b_ih_n

    // ---- gates: acc0[j] = r_j (half0 lanes) / z_j (half1 lanes), same batch n
#pragma unroll
    for (int j = 0; j < 8; ++j) {
      float sig = fast_sigmoid(acc0[j]);        // lo: r_j(b); hi: z_j(b)
      float sw  = x16(sig);                     // the other gate, same batch
      float rv  = lo ? sig : sw;                // r_j(b) on all lanes
      float zv  = lo ? sw : sig;                // z_j(b) on all lanes
      float pre = fmaf(rv, acc1[j], acc2[j]);   // gx_n + r*gh_n (all lanes)
      float nv  = fast_tanh(pre);
      float hn  = fmaf(zv, hreg[j] - nv, nv);   // (1-z)*n + z*h
      hreg[j] = hn;                             // valid in all 32 lanes
    }

    // ---- ONE fully-active b128 store: half0 -> h[0:4], half1 -> h[4:8]
    v4f o;
    o.x = lo ? hreg[0] : hreg[4];
    o.y = lo ? hreg[1] : hreg[5];
    o.z = lo ? hreg[2] : hreg[6];
    o.w = lo ? hreg[3] : hreg[7];
    __builtin_nontemporal_store(o, (v4f*)(out + obase + t * (NF * H)));

    xcur = xnext;
  }
}

extern "C" void kernel_launch(void* const* d_in, const int* in_sizes, int n_in,
                              void* d_out, int out_size, void* d_ws, size_t ws_size,
                              hipStream_t stream) {
  const float* x   = (const float*)d_in[0];  // [BS,TS,NF]
  const float* wih = (const float*)d_in[1];  // [NF,24,1]
  const float* whh = (const float*)d_in[2];  // [NF,24,8]
  const float* bi  = (const float*)d_in[3];  // [NF,24]
  const float* bh  = (const float*)d_in[4];  // [NF,24]
  float* out = (float*)d_out;                // [BS,TS,NF,H]

  dim3 grid(NF, BS / 16);  // 73 features x 8 batch slabs = 584 waves
  mcgru_wmma_kernel<<<grid, 32, 0, stream>>>(x, wih, whh, bi, bh, out);
}